// GaussianRenderer_64725157151169
// MI455X (gfx1250) — compile-verified
//
#include <hip/hip_runtime.h>
#include <math.h>

typedef float v2f __attribute__((ext_vector_type(2)));
typedef float v8f __attribute__((ext_vector_type(8)));

// ---------------------------------------------------------------------------
// Kernel 1: per-gaussian preprocessing.
// Writes z to zbuf and 16 floats per gaussian to params:
//   [0..7]  quad coeffs for dist(x,y) = c0*x^2 + c1*xy + c2*y^2 + c3*x + c4*y + c5  (c6=c7=0)
//   [8..11] bbox xm, ym, xM, yM
//   [12]    effective opacity (0 if z<=0)
//   [13..15] color rgb
// ---------------------------------------------------------------------------
__global__ void gs_prep(const float* __restrict__ means,
                        const float* __restrict__ covs,
                        const float* __restrict__ colors,
                        const float* __restrict__ opac,
                        const float* __restrict__ Rm,
                        const float* __restrict__ tv,
                        const int* __restrict__ pfx, const int* __restrict__ pfy,
                        const int* __restrict__ pcx, const int* __restrict__ pcy,
                        const int* __restrict__ pw,  const int* __restrict__ ph,
                        float* __restrict__ zbuf, float* __restrict__ params, int N)
{
    int g = blockIdx.x * blockDim.x + threadIdx.x;
    if (g >= N) return;

    const float fx = (float)pfx[0], fy = (float)pfy[0];
    const float cx = (float)pcx[0], cy = (float)pcy[0];
    const float Wf = (float)pw[0],  Hf = (float)ph[0];

    float R[3][3];
#pragma unroll
    for (int i = 0; i < 3; ++i)
#pragma unroll
        for (int j = 0; j < 3; ++j) R[i][j] = Rm[i * 3 + j];

    const float m0 = means[g * 3 + 0], m1 = means[g * 3 + 1], m2 = means[g * 3 + 2];
    const float camX = R[0][0] * m0 + R[0][1] * m1 + R[0][2] * m2 + tv[0];
    const float camY = R[1][0] * m0 + R[1][1] * m1 + R[1][2] * m2 + tv[1];
    const float camZ = R[2][0] * m0 + R[2][1] * m1 + R[2][2] * m2 + tv[2];
    const float z = camZ;
    const float invz = 1.0f / z;
    const float mx = camX * invz * fx + cx;
    const float my = camY * invz * fy + cy;

    // cov_cam = R * S * R^T
    const float* S = covs + (size_t)g * 9;
    float M[3][3], C[3][3];
#pragma unroll
    for (int i = 0; i < 3; ++i)
#pragma unroll
        for (int j = 0; j < 3; ++j)
            M[i][j] = R[i][0] * S[0 * 3 + j] + R[i][1] * S[1 * 3 + j] + R[i][2] * S[2 * 3 + j];
#pragma unroll
    for (int i = 0; i < 3; ++i)
#pragma unroll
        for (int j = 0; j < 3; ++j)
            C[i][j] = M[i][0] * R[j][0] + M[i][1] * R[j][1] + M[i][2] * R[j][2];

    // J (2x3)
    const float J0x = fx * invz, J0y = 0.0f, J0z = -fx * camX * invz * invz;
    const float J1x = 0.0f, J1y = fy * invz, J1z = -fy * camY * invz * invz;

    // cov2d = J * C * J^T
    float t0[3], t1[3];
#pragma unroll
    for (int i = 0; i < 3; ++i) {
        t0[i] = C[i][0] * J0x + C[i][1] * J0y + C[i][2] * J0z;
        t1[i] = C[i][0] * J1x + C[i][1] * J1y + C[i][2] * J1z;
    }
    const float a  = J0x * t0[0] + J0y * t0[1] + J0z * t0[2];
    const float b  = J0x * t1[0] + J0y * t1[1] + J0z * t1[2];
    const float c2 = J1x * t0[0] + J1y * t0[1] + J1z * t0[2];
    const float d  = J1x * t1[0] + J1y * t1[1] + J1z * t1[2];

    // 2x2 symmetric eigen (eigh, ascending eigenvalues)
    const float bs   = 0.5f * (b + c2);
    const float mean = 0.5f * (a + d);
    const float disc = sqrtf(0.25f * (a - d) * (a - d) + bs * bs);
    const float l0 = mean - disc;
    const float l1 = mean + disc;
    float v0x = bs, v0y = l0 - a;
    float n2 = v0x * v0x + v0y * v0y;
    if (n2 < 1e-30f) { v0x = (a <= d) ? 1.0f : 0.0f; v0y = (a <= d) ? 0.0f : 1.0f; n2 = 1.0f; }
    const float inr = rsqrtf(n2);
    v0x *= inr; v0y *= inr;
    const float v1x = -v0y, v1y = v0x;
    const float s0 = 3.0f * sqrtf(fabsf(l0));
    const float s1 = 3.0f * sqrtf(fabsf(l1));
    const float ex = fmaxf(fabsf(s0 * v0x), fabsf(s1 * v1x));
    const float ey = fmaxf(fabsf(s0 * v0y), fabsf(s1 * v1y));

    const float xm = fmaxf(0.0f, floorf(mx - ex));
    const float ym = fmaxf(0.0f, floorf(my - ey));
    const float xM = fminf(Wf, ceilf(mx + ex));
    const float yM = fminf(Hf, ceilf(my + ey));

    // inverse covariance and quadratic expansion about (mx,my)
    const float det = a * d - b * c2;
    const float idet = 1.0f / det;
    const float ia  = d * idet;
    const float ib  = -b * idet;
    const float ic  = -c2 * idet;
    const float idd = a * idet;
    const float Bq  = ib + ic;
    const float c3 = -(2.0f * ia * mx + Bq * my);
    const float c4 = -(2.0f * idd * my + Bq * mx);
    const float c5 = ia * mx * mx + Bq * mx * my + idd * my * my;

    float opv = opac[g];
    if (!(z > 0.0f)) opv = 0.0f;

    float* p = params + (size_t)g * 16;
    p[0] = ia;  p[1] = Bq;  p[2] = idd; p[3] = c3;
    p[4] = c4;  p[5] = c5;  p[6] = 0.0f; p[7] = 0.0f;
    p[8] = xm;  p[9] = ym;  p[10] = xM; p[11] = yM;
    p[12] = opv;
    p[13] = colors[g * 3 + 0];
    p[14] = colors[g * 3 + 1];
    p[15] = colors[g * 3 + 2];
    zbuf[g] = z;
}

// ---------------------------------------------------------------------------
// Kernel 2: stable rank sort by descending z (matches jnp.argsort(-z)) + scatter.
// ---------------------------------------------------------------------------
__global__ void gs_sort_scatter(const float* __restrict__ zbuf,
                                const float* __restrict__ params,
                                float* __restrict__ sorted, int N)
{
    int g = blockIdx.x * blockDim.x + threadIdx.x;
    if (g >= N) return;
    const float zg = zbuf[g];
    int rank = 0;
    for (int j = 0; j < N; ++j) {
        const float zj = zbuf[j];
        rank += (zj > zg) || (zj == zg && j < g);
    }
    const float* src = params + (size_t)g * 16;
    float* dst = sorted + (size_t)rank * 16;
#pragma unroll
    for (int k = 0; k < 16; ++k) dst[k] = src[k];
}

// ---------------------------------------------------------------------------
// Kernel 3: per-group (16 gaussians) union bbox for wave-uniform group skip.
// ---------------------------------------------------------------------------
__global__ void gs_group_bbox(const float* __restrict__ sorted,
                              float* __restrict__ gbb, int ngroups)
{
    int grp = blockIdx.x * blockDim.x + threadIdx.x;
    if (grp >= ngroups) return;
    float xm = 1e30f, ym = 1e30f, xM = -1e30f, yM = -1e30f;
    for (int k = 0; k < 16; ++k) {
        const float* p = sorted + (size_t)(grp * 16 + k) * 16;
        xm = fminf(xm, p[8]);
        ym = fminf(ym, p[9]);
        xM = fmaxf(xM, p[10]);
        yM = fmaxf(yM, p[11]);
    }
    gbb[grp * 4 + 0] = xm;
    gbb[grp * 4 + 1] = ym;
    gbb[grp * 4 + 2] = xM;
    gbb[grp * 4 + 3] = yM;
}

// ---------------------------------------------------------------------------
// Kernel 4: render. One wave per 16-pixel horizontal strip.
// dist[16 gauss x 16 px] via two chained V_WMMA_F32_16X16X4_F32 (K=8 features).
// A (16x4, M=gaussian): lanes 0-15 -> M=lane, K={0,1}; lanes 16-31 -> K={2,3}.
// B (4x16, N=pixel):    lanes 0-15 -> N=lane, K={0,1}; lanes 16-31 -> K={2,3}.
// D (16x16 f32): lane<16 -> pixel=lane, gaussians 0-7 in 8 VGPRs;
//                lane>=16 -> pixel=lane-16, gaussians 8-15.
// Each half composits its 8 gaussians, halves merged with one shfl_xor(16).
// ---------------------------------------------------------------------------
__global__ __launch_bounds__(256)
void gs_render(const float* __restrict__ sorted,
               const float* __restrict__ gbb,
               float* __restrict__ out, int N, int Wi)
{
    const int lane  = threadIdx.x & 31;
    const int wave  = threadIdx.x >> 5;
    const int strip = blockIdx.x * (blockDim.x >> 5) + wave;
    const int spr   = Wi >> 4;                 // strips per row
    const int py    = strip / spr;
    const int x0    = (strip - py * spr) << 4;
    const int c16   = lane & 15;
    const int half  = lane >> 4;
    const float fpx = (float)(x0 + c16);
    const float fpy = (float)py;
    const float x0f = (float)x0;
    const float x1f = (float)(x0 + 15);

    // Feature matrix B: K rows = [x^2, xy, y^2, x | y, 1, 0, 0]
    v2f B0, B1;
    B0.x = half ? fpy * fpy : fpx * fpx;
    B0.y = half ? fpx       : fpx * fpy;
    B1.x = half ? 0.0f      : fpy;
    B1.y = half ? 0.0f      : 1.0f;
    const int k0 = half << 1;

    float imgR = 0.0f, imgG = 0.0f, imgB = 0.0f;

    for (int base = 0; base < N; base += 16) {
        const float4 gb = *(const float4*)(gbb + (size_t)(base >> 4) * 4);
        const bool inter = (fpy >= gb.y) && (fpy < gb.w) && (x1f >= gb.x) && (x0f < gb.z);
        if (__ballot(inter) == 0ull) continue;   // uniform branch: EXEC stays all-1s

        // A: coeffs of gaussian (base + c16), K slice per half
        const float* cp = sorted + (size_t)(base + c16) * 16;
        v2f A0 = *(const v2f*)(cp + k0);
        v2f A1 = *(const v2f*)(cp + k0 + 4);

        v8f acc = 0.0f;
        acc = __builtin_amdgcn_wmma_f32_16x16x4_f32(false, A0, false, B0, (short)0, acc, false, false);
        acc = __builtin_amdgcn_wmma_f32_16x16x4_f32(false, A1, false, B1, (short)0, acc, false, false);

        // composite this half's 8 gaussians over transparent background
        float pR = 0.0f, pG = 0.0f, pB = 0.0f, pT = 1.0f;
#pragma unroll
        for (int r = 0; r < 8; ++r) {
            const float* pp = sorted + (size_t)(base + (half << 3) + r) * 16;
            const float4 bb = *(const float4*)(pp + 8);
            const float4 oc = *(const float4*)(pp + 12);
            const float w = __expf(-0.5f * acc[r]);
            const bool valid = (fpx >= bb.x) && (fpx < bb.z) && (fpy >= bb.y) && (fpy < bb.w);
            const float alpha = valid ? w * oc.x : 0.0f;
            const float om = 1.0f - alpha;
            pR = oc.y * alpha + pR * om;
            pG = oc.z * alpha + pG * om;
            pB = oc.w * alpha + pB * om;
            pT *= om;
        }

        // merge halves: group = (gaussians 8-15) over (gaussians 0-7)
        const float oR = __shfl_xor(pR, 16);
        const float oG = __shfl_xor(pG, 16);
        const float oB = __shfl_xor(pB, 16);
        const float oT = __shfl_xor(pT, 16);
        const float aR = half ? oR : pR, aG = half ? oG : pG, aB = half ? oB : pB;
        const float bR = half ? pR : oR, bG = half ? pG : oG, bB = half ? pB : oB;
        const float bT = half ? pT : oT;
        const float gT = pT * oT;
        imgR = (bR + bT * aR) + gT * imgR;
        imgG = (bG + bT * aG) + gT * imgG;
        imgB = (bB + bT * aB) + gT * imgB;
    }

    if (half == 0) {
        float* o = out + (size_t)(py * Wi + x0 + c16) * 3;
        o[0] = imgR; o[1] = imgG; o[2] = imgB;
    }
}

// ---------------------------------------------------------------------------
extern "C" void kernel_launch(void* const* d_in, const int* in_sizes, int n_in,
                              void* d_out, int out_size, void* d_ws, size_t ws_size,
                              hipStream_t stream)
{
    const float* means  = (const float*)d_in[0];
    const float* covs   = (const float*)d_in[1];
    const float* colors = (const float*)d_in[2];
    const float* opac   = (const float*)d_in[3];
    const float* Rm     = (const float*)d_in[4];
    const float* tv     = (const float*)d_in[5];
    const int* pfx = (const int*)d_in[6];
    const int* pfy = (const int*)d_in[7];
    const int* pcx = (const int*)d_in[8];
    const int* pcy = (const int*)d_in[9];
    const int* pw  = (const int*)d_in[10];
    const int* ph  = (const int*)d_in[11];

    const int N = in_sizes[0] / 3;          // 1024
    const int W = 256, H = 256;             // fixed by reference

    float* ws     = (float*)d_ws;
    float* zbuf   = ws;                     // N
    float* params = zbuf + N;               // N*16
    float* sorted = params + (size_t)N * 16;// N*16
    float* gbb    = sorted + (size_t)N * 16;// (N/16)*4
    float* out    = (float*)d_out;

    gs_prep<<<(N + 255) / 256, 256, 0, stream>>>(means, covs, colors, opac, Rm, tv,
                                                 pfx, pfy, pcx, pcy, pw, ph,
                                                 zbuf, params, N);
    gs_sort_scatter<<<(N + 255) / 256, 256, 0, stream>>>(zbuf, params, sorted, N);
    gs_group_bbox<<<1, 64, 0, stream>>>(sorted, gbb, N / 16);

    const int strips = H * (W / 16);        // 4096 waves
    gs_render<<<strips / 8, 256, 0, stream>>>(sorted, gbb, out, N, W);
}